// PrioriLoss_15668040696555
// MI455X (gfx1250) — compile-verified
//
#include <hip/hip_runtime.h>

typedef __attribute__((ext_vector_type(2))) float v2f;
typedef __attribute__((ext_vector_type(4))) float v4f;
typedef __attribute__((ext_vector_type(8))) float v8f;

#define MIN_V 20.0f
#define MAX_V 80.0f
#define SCALE 2.0f

#define NBLK 1024
#define TPB  256

// ---------------------------------------------------------------------------
// Kernel 1: streaming weighted squared-error partial sums.
// 128 MiB one-shot stream @ 23.3 TB/s HBM -> ~5.8 us roofline. B128
// non-temporal loads, counted loop + unroll 4 => 8 independent B128 loads in
// flight per wave, 8192 wave32s for occupancy. fp32 accumulation (only ~64
// elements per thread before the tree reduce -> no precision concern).
// ---------------------------------------------------------------------------
__global__ __launch_bounds__(TPB) void wloss_partial(const float* __restrict__ pred,
                                                     const float* __restrict__ tru,
                                                     float* __restrict__ partial,
                                                     int n) {
    const int tid    = blockIdx.x * blockDim.x + threadIdx.x;
    const int stride = gridDim.x * blockDim.x;
    const int n4     = n >> 2;

    const v4f* __restrict__ p4 = (const v4f*)pred;
    const v4f* __restrict__ t4 = (const v4f*)tru;

    const int iters = n4 / stride;          // exact for N=2^24: 16
    const int rem   = n4 - iters * stride;

    float acc = 0.0f;
    int   i   = tid;
#pragma unroll 4
    for (int k = 0; k < iters; ++k, i += stride) {
        v4f p = __builtin_nontemporal_load(p4 + i);
        v4f t = __builtin_nontemporal_load(t4 + i);
#pragma unroll
        for (int c = 0; c < 4; ++c) {
            float pe = p[c];
            float d  = pe - t[c];
            float w  = (pe < MIN_V || pe > MAX_V) ? SCALE : 1.0f;
            acc = fmaf(w * d, d, acc);
        }
    }
    if (tid < rem) {                        // leftover v4f chunks
        v4f p = __builtin_nontemporal_load(p4 + i);
        v4f t = __builtin_nontemporal_load(t4 + i);
#pragma unroll
        for (int c = 0; c < 4; ++c) {
            float pe = p[c];
            float d  = pe - t[c];
            float w  = (pe < MIN_V || pe > MAX_V) ? SCALE : 1.0f;
            acc = fmaf(w * d, d, acc);
        }
    }
    const int base = n4 << 2;               // scalar tail (n % 4)
    if (tid < (n - base)) {
        float pe = pred[base + tid];
        float d  = pe - tru[base + tid];
        float w  = (pe < MIN_V || pe > MAX_V) ? SCALE : 1.0f;
        acc = fmaf(w * d, d, acc);
    }

    // wave32 shuffle reduction
#pragma unroll
    for (int off = 16; off > 0; off >>= 1)
        acc += __shfl_xor(acc, off, 32);

    __shared__ float smem[TPB / 32];
    const int lane = threadIdx.x & 31;
    const int wid  = threadIdx.x >> 5;
    if (lane == 0) smem[wid] = acc;
    __syncthreads();

    if (threadIdx.x == 0) {
        float s = 0.0f;
#pragma unroll
        for (int w = 0; w < TPB / 32; ++w) s += smem[w];
        partial[blockIdx.x] = s;
    }
}

// ---------------------------------------------------------------------------
// Kernel 2: single-wave final reduce. NBLK partials loaded as fully-unrolled
// independent B128 loads (all in flight before one wait), then a
// V_WMMA_F32_16X16X4_F32 collapses the 32 lanes:
//   A (16x4 f32 layout): lanes 0-15 put value at K=0, lanes 16-31 at K=2.
//   B = all-ones        -> D[m][n] = sum_k A[m][k] = lane_m + lane_{m+16}.
//   C/D layout: lane l's 8 VGPRs hold rows {0..7} (l<16) or {8..15} (l>=16),
//   so sum(d[0..7]) in lane 0 + same value from lane 16 = total of all lanes.
// Exactly one wave (32 threads) so EXEC is all-1s for WMMA. inv_n is
// precomputed on the host (exact: N is a power of two) to avoid the runtime
// fp32 divide sequence.
// ---------------------------------------------------------------------------
__global__ __launch_bounds__(32) void wloss_final(const float* __restrict__ partial,
                                                  float* __restrict__ out,
                                                  float inv_n) {
    const int lane = threadIdx.x;
    const v4f* __restrict__ p4 = (const v4f*)partial;   // NBLK/4 chunks

    float s = 0.0f;
#pragma unroll
    for (int k = 0; k < NBLK / 128; ++k) {              // 8 independent B128 loads
        v4f v = p4[lane + k * 32];
        s += (v[0] + v[1]) + (v[2] + v[3]);
    }

    v2f a; a[0] = s;    a[1] = 0.0f;
    v2f b; b[0] = 1.0f; b[1] = 1.0f;
    v8f c = {};
    // 8 args: (neg_a, A, neg_b, B, c_mod, C, reuse_a, reuse_b)
    v8f d = __builtin_amdgcn_wmma_f32_16x16x4_f32(
        false, a, false, b, (short)0, c, false, false);

    float t = ((d[0] + d[1]) + (d[2] + d[3])) + ((d[4] + d[5]) + (d[6] + d[7]));
    float hi = __shfl(t, 16, 32);   // rows 8..15 live in lane 16
    if (lane == 0) out[0] = (t + hi) * inv_n;
}

extern "C" void kernel_launch(void* const* d_in, const int* in_sizes, int n_in,
                              void* d_out, int out_size, void* d_ws, size_t ws_size,
                              hipStream_t stream) {
    const float* pred = (const float*)d_in[0];
    const float* tru  = (const float*)d_in[1];
    float*       out  = (float*)d_out;
    float*       part = (float*)d_ws;   // NBLK floats = 4 KiB scratch
    const int    n    = in_sizes[0];
    const float  inv_n = 1.0f / (float)n;

    wloss_partial<<<NBLK, TPB, 0, stream>>>(pred, tru, part, n);
    wloss_final<<<1, 32, 0, stream>>>(part, out, inv_n);
}